// Model_86517821215868
// MI455X (gfx1250) — compile-verified
//
#include <hip/hip_runtime.h>
#include <hip/hip_bf16.h>
#include <stdint.h>

// ---- problem constants (match reference) ----
#define CONF_T  0.25f
#define IOU_T   0.45f
#define MAX_WHF 4096.0f
static constexpr int Bb = 16, Nn = 25200, NCc = 80;
static constexpr int TOPK = 2048, MAXDET = 300, CAP = 4096;   // CAP = compaction buffer (>= TOPK + max bin pop)

typedef __attribute__((ext_vector_type(2))) float v2f;
typedef __attribute__((ext_vector_type(8))) float v8f;

// async-to-LDS builtin operand types (per hipcc diagnostic: vector_size(16) int,
// global AS for the source side)
typedef int v4i_vs __attribute__((vector_size(16)));
typedef __attribute__((address_space(1))) v4i_vs gv4i;
typedef __attribute__((address_space(3))) v4i_vs lv4i;

#if defined(__has_builtin)
#if __has_builtin(__builtin_amdgcn_wmma_f32_16x16x4_f32)
#define HAVE_WMMA_F32X4 1
#endif
#if __has_builtin(__builtin_amdgcn_ballot_w32)
#define HAVE_BALLOT32 1
#endif
#if __has_builtin(__builtin_amdgcn_global_load_async_to_lds_b128) && \
    __has_builtin(__builtin_amdgcn_s_wait_asynccnt)
#define HAVE_ASYNC_LDS 1
#endif
#endif

// ------------------------------------------------------------------
// Kernel A: streaming pass over 137MB of predictions -> score[B*N]
// score = (obj>0.25 && obj*maxcls>0.25) ? obj*maxcls : 0
// ------------------------------------------------------------------
__global__ __launch_bounds__(256) void score_kernel(const float* __restrict__ pred,
                                                    float* __restrict__ score) {
    int idx = blockIdx.x * 256 + threadIdx.x;
    if (idx >= Bb * Nn) return;
    const float* p = pred + (size_t)idx * 85;
    __builtin_prefetch(p + 85 * 256, 0, 0);   // global_prefetch_b8 (next tile this thread touches)
    float obj = p[4];
    float m = p[5];
#pragma unroll 4
    for (int c = 1; c < NCc; ++c) m = fmaxf(m, p[5 + c]);
    float conf = obj * m;
    bool valid = (obj > CONF_T) && (conf > CONF_T);
    score[idx] = valid ? conf : 0.f;
}

// ------------------------------------------------------------------
// Kernel B: per-batch top-2048. Histogram threshold -> compact into
// LDS -> bitonic sort 4096 u64 keys (score|~index, descending) ->
// gather boxes/cls for the top 2048.
// ------------------------------------------------------------------
__global__ __launch_bounds__(1024) void topk_kernel(const float* __restrict__ pred,
                                                    const float* __restrict__ score,
                                                    float* __restrict__ topscore,
                                                    float* __restrict__ topcls,
                                                    float4* __restrict__ topbox,
                                                    float4* __restrict__ topobox) {
    __shared__ unsigned hist[1024];
    __shared__ unsigned long long keys[CAP];
    __shared__ int scnt;
    __shared__ int tbin;

    const int b = blockIdx.x, tid = threadIdx.x;
    if (tid == 0) scnt = 0;
    hist[tid] = 0;
    __syncthreads();

    const float* sc = score + (size_t)b * Nn;
    for (int n = tid; n < Nn; n += 1024) {
        float s = sc[n];
        if (s > CONF_T) {
            int bin = min((int)(s * 1024.f), 1023);
            atomicAdd(&hist[bin], 1u);
        }
    }
    __syncthreads();

    if (tid == 0) {
        int cum = 0, bsel = 256;                 // bin 256 edge ~= 0.25
        for (int bb2 = 1023; bb2 >= 256; --bb2) {
            cum += (int)hist[bb2];
            bsel = bb2;
            if (cum >= TOPK) break;
        }
        tbin = bsel;
    }
    __syncthreads();

    const int t = tbin;
    for (int n = tid; n < Nn; n += 1024) {
        float s = sc[n];
        if (s > CONF_T) {
            int bin = min((int)(s * 1024.f), 1023);
            if (bin >= t) {
                int pos = atomicAdd(&scnt, 1);
                if (pos < CAP)
                    keys[pos] = ((unsigned long long)__float_as_uint(s) << 32) |
                                (unsigned long long)(0xFFFFFFFFu - (unsigned)n);
            }
        }
    }
    __syncthreads();
    const int total = min(scnt, CAP);
    for (int i = tid; i < CAP; i += 1024)
        if (i >= total) keys[i] = 0ull;
    __syncthreads();

    // bitonic sort, descending (key = score bits | ~index -> stable like lax.top_k)
    for (int k2 = 2; k2 <= CAP; k2 <<= 1) {
        for (int j = k2 >> 1; j > 0; j >>= 1) {
            for (int i = tid; i < CAP; i += 1024) {
                int ixj = i ^ j;
                if (ixj > i) {
                    unsigned long long a = keys[i], bk = keys[ixj];
                    bool up = ((i & k2) == 0);
                    if (up ? (a < bk) : (a > bk)) { keys[i] = bk; keys[ixj] = a; }
                }
            }
            __syncthreads();
        }
    }

    // emit top-2048 + gather boxes (re-read prediction rows only for winners)
    for (int k = tid; k < TOPK; k += 1024) {
        unsigned long long key = keys[k];
        float s = __uint_as_float((unsigned)(key >> 32));
        unsigned n = 0xFFFFFFFFu - (unsigned)(key & 0xFFFFFFFFull);
        size_t o = (size_t)b * TOPK + k;
        if (s > CONF_T && n < (unsigned)Nn) {
            const float* p = pred + ((size_t)b * Nn + n) * 85;
            int bj = 0; float bc = p[5];
            for (int c = 1; c < NCc; ++c) { float v = p[5 + c]; if (v > bc) { bc = v; bj = c; } }
            float x = p[0], y = p[1], w = p[2] * 0.5f, h = p[3] * 0.5f;
            float4 box = make_float4(x - w, y - h, x + w, y + h);
            float offc = (float)bj * MAX_WHF;
            topscore[o] = s;
            topcls[o]   = (float)bj;
            topbox[o]   = box;
            topobox[o]  = make_float4(box.x + offc, box.y + offc, box.z + offc, box.w + offc);
        } else {
            topscore[o] = 0.f; topcls[o] = 0.f;
            topbox[o]  = make_float4(0.f, 0.f, 0.f, 0.f);
            topobox[o] = make_float4(0.f, 0.f, 0.f, 0.f);
        }
    }
}

// ------------------------------------------------------------------
// Kernel C: 2048x2048 IoU > thresh bitmask per batch.
// One block = 16 rows; 8 waves sweep 16-col tiles.
//  - boxes staged to LDS with global_load_async_to_lds_b128 (ASYNCcnt)
//  - pairwise area sum a1[i]+a2[j] per 16x16 tile via
//    V_WMMA_F32_16X16X4_F32 (A = [a1 | 1 | 0 | 0], B = [1 ; a2 ; 0 ; 0])
//  - IoU test as inter > IOU_T*union (no divide pipeline)
//  - 8 wave32 ballots per tile, then only two exec transitions for stores
// ------------------------------------------------------------------
__global__ __launch_bounds__(256) void iou_mask_kernel(const float4* __restrict__ topobox,
                                                       unsigned short* __restrict__ mask) {
    __shared__ float4 sB[TOPK];
    __shared__ float  sA[TOPK];
    const int b = blockIdx.x >> 7;
    const int rowTile = blockIdx.x & 127;
    const int tid = threadIdx.x;
    const float4* ob = topobox + (size_t)b * TOPK;

#if defined(HAVE_ASYNC_LDS)
    // async copy: 2048 float4 = 32KB, 8 x b128 per thread; then areas from LDS
    for (int i = tid; i < TOPK; i += 256) {
        gv4i* g = (gv4i*)(uintptr_t)(ob + i);
        lv4i* l = (lv4i*)(uint32_t)(uintptr_t)(sB + i);
        __builtin_amdgcn_global_load_async_to_lds_b128(g, l, 0, 0);
    }
    __builtin_amdgcn_s_wait_asynccnt(0);
    __syncthreads();
    for (int i = tid; i < TOPK; i += 256) {
        float4 v = sB[i];
        sA[i] = (v.z - v.x) * (v.w - v.y);
    }
#else
    for (int i = tid; i < TOPK; i += 256) {
        float4 v = ob[i];
        sB[i] = v;
        sA[i] = (v.z - v.x) * (v.w - v.y);
    }
#endif
    __syncthreads();

    const int lane = tid & 31, wid = tid >> 5;
    const int rowbase = rowTile * 16;

    // A operand: lanes 0-15 -> {K0 = a1[row], K1 = 1}; lanes 16-31 -> {K2,K3} = 0
    v2f A;
    A.x = (lane < 16) ? sA[rowbase + lane] : 0.f;
    A.y = (lane < 16) ? 1.0f : 0.f;

    for (int ct = wid; ct < 128; ct += 8) {
        const int colbase = ct * 16;
        // B operand: lanes 0-15 -> {K0 row = 1, K1 row = a2[col]}; lanes 16-31 zero
        v2f Bv;
        Bv.x = (lane < 16) ? 1.0f : 0.f;
        Bv.y = (lane < 16) ? sA[colbase + lane] : 0.f;

        float pairsum[8];
#if defined(HAVE_WMMA_F32X4)
        v8f c = {};
        v8f d = __builtin_amdgcn_wmma_f32_16x16x4_f32(false, A, false, Bv,
                                                      (short)0, c, false, false);
#pragma unroll
        for (int r = 0; r < 8; ++r) pairsum[r] = d[r];
#else
#pragma unroll
        for (int r = 0; r < 8; ++r) {
            int M = r + ((lane >> 4) << 3);
            pairsum[r] = sA[rowbase + M] + sA[colbase + (lane & 15)];
        }
#endif
        // All 8 ballots with full EXEC (no divergence yet in this iteration)
        unsigned bals[8];
#pragma unroll
        for (int r = 0; r < 8; ++r) {
            // D layout: VGPR r, lanes 0-15 -> M=r; lanes 16-31 -> M=r+8; N = lane%16
            const int M = r + ((lane >> 4) << 3);
            const int i = rowbase + M;
            const int j = colbase + (lane & 15);
            float4 b1 = sB[i], b2 = sB[j];
            float lx = fmaxf(b1.x, b2.x), ly = fmaxf(b1.y, b2.y);
            float rx = fminf(b1.z, b2.z), ry = fminf(b1.w, b2.w);
            float inter = fmaxf(rx - lx, 0.f) * fmaxf(ry - ly, 0.f);
            // iou > T  <=>  inter > T*(a1+a2-inter)   (union >= 0; padded rows: 0>0 false)
            bool pr = inter > IOU_T * (pairsum[r] - inter);
#if defined(HAVE_BALLOT32)
            bals[r] = __builtin_amdgcn_ballot_w32(pr);
#else
            bals[r] = (unsigned)__ballot(pr);
#endif
        }
        // Two exec transitions total: lane 0 stores rows r, lane 16 stores rows r+8
        unsigned short* mrowp = mask + ((size_t)b * TOPK + rowbase) * 128 + ct;
        if (lane == 0) {
#pragma unroll
            for (int r = 0; r < 8; ++r)
                mrowp[(size_t)r * 128] = (unsigned short)(bals[r] & 0xFFFFu);
        } else if (lane == 16) {
#pragma unroll
            for (int r = 0; r < 8; ++r)
                mrowp[(size_t)(r + 8) * 128] = (unsigned short)(bals[r] >> 16);
        }
    }
}

// ------------------------------------------------------------------
// Kernel D: sequential greedy suppression (reference scan semantics).
// 128 threads each own one 16-bit word of the removed/keep bitmaps.
// ------------------------------------------------------------------
__global__ __launch_bounds__(128) void nms_scan_kernel(const float* __restrict__ topscore,
                                                       const unsigned short* __restrict__ mask,
                                                       unsigned short* __restrict__ keepw) {
    __shared__ int decision;
    __shared__ float ss[TOPK];
    const int b = blockIdx.x, tid = threadIdx.x;
    for (int i = tid; i < TOPK; i += 128) ss[i] = topscore[(size_t)b * TOPK + i];
    __syncthreads();

    unsigned removed = 0, keepbits = 0;
    const unsigned short* mrow = mask + (size_t)b * TOPK * 128;
    for (int i = 0; i < TOPK; ++i) {
        const int owner = i >> 4, bit = i & 15;
        if (tid == owner) {
            int ki = (ss[i] > CONF_T) && !((removed >> bit) & 1u);
            if (ki) keepbits |= (1u << bit);
            decision = ki;
        }
        __syncthreads();
        if (decision) removed |= (unsigned)mrow[(size_t)i * 128 + tid];
        __syncthreads();
    }
    keepw[b * 128 + tid] = (unsigned short)keepbits;
}

// ------------------------------------------------------------------
// Kernel E: compact kept boxes (already score-descending) into the
// 300x6 output + count. Fully overwrites the poisoned d_out.
// ------------------------------------------------------------------
__global__ __launch_bounds__(256) void output_kernel(const float* __restrict__ topscore,
                                                     const float* __restrict__ topcls,
                                                     const float4* __restrict__ topbox,
                                                     const unsigned short* __restrict__ keepw,
                                                     float* __restrict__ out) {
    __shared__ unsigned kw[128];
    __shared__ int wpre[129];
    const int b = blockIdx.x, tid = threadIdx.x;
    if (tid < 128) kw[tid] = keepw[b * 128 + tid];
    __syncthreads();
    if (tid == 0) {
        int run = 0;
        for (int w = 0; w < 128; ++w) { wpre[w] = run; run += __popc(kw[w]); }
        wpre[128] = run;
    }
    __syncthreads();

    float* det = out + (size_t)b * MAXDET * 6;
    for (int i = tid; i < MAXDET * 6; i += 256) det[i] = 0.f;
    __syncthreads();

    for (int i = tid; i < TOPK; i += 256) {
        unsigned w = kw[i >> 4];
        int bit = i & 15;
        if ((w >> bit) & 1u) {
            int rank = wpre[i >> 4] + __popc(w & ((1u << bit) - 1u));
            if (rank < MAXDET) {
                float4 bx = topbox[(size_t)b * TOPK + i];
                float* row = det + (size_t)rank * 6;
                row[0] = bx.x; row[1] = bx.y; row[2] = bx.z; row[3] = bx.w;
                row[4] = topscore[(size_t)b * TOPK + i];
                row[5] = topcls[(size_t)b * TOPK + i];
            }
        }
    }
    if (tid == 0) out[(size_t)Bb * MAXDET * 6 + b] = (float)wpre[128];
}

// ------------------------------------------------------------------
extern "C" void kernel_launch(void* const* d_in, const int* in_sizes, int n_in,
                              void* d_out, int out_size, void* d_ws, size_t ws_size,
                              hipStream_t stream) {
    (void)in_sizes; (void)n_in; (void)out_size; (void)ws_size;
    const float* pred = (const float*)d_in[0];
    float* out = (float*)d_out;

    char* ws = (char*)d_ws;
    size_t off = 0;
    auto alloc = [&](size_t bytes) -> char* {
        char* p = ws + off;
        off = (off + bytes + 255) & ~(size_t)255;
        return p;
    };
    float*  score    = (float*)alloc((size_t)Bb * Nn * sizeof(float));        // 1.6 MB
    float*  topscore = (float*)alloc((size_t)Bb * TOPK * sizeof(float));
    float*  topcls   = (float*)alloc((size_t)Bb * TOPK * sizeof(float));
    float4* topbox   = (float4*)alloc((size_t)Bb * TOPK * sizeof(float4));
    float4* topobox  = (float4*)alloc((size_t)Bb * TOPK * sizeof(float4));
    unsigned short* mask  = (unsigned short*)alloc((size_t)Bb * TOPK * 128 * 2); // 8 MB
    unsigned short* keepw = (unsigned short*)alloc((size_t)Bb * 128 * 2);
    // total workspace ~11.3 MB

    score_kernel<<<(Bb * Nn + 255) / 256, 256, 0, stream>>>(pred, score);
    topk_kernel<<<Bb, 1024, 0, stream>>>(pred, score, topscore, topcls, topbox, topobox);
    iou_mask_kernel<<<Bb * 128, 256, 0, stream>>>(topobox, mask);
    nms_scan_kernel<<<Bb, 128, 0, stream>>>(topscore, mask, keepw);
    output_kernel<<<Bb, 256, 0, stream>>>(topscore, topcls, topbox, keepw, out);
}